// VoiceDDSP_RealTime_63565515981201
// MI455X (gfx1250) — compile-verified
//
#include <hip/hip_runtime.h>
#include <cstdint>

// ---------------- problem constants ----------------
#define SR_F 24000.0f
#define PI_F 3.14159265358979323846f

constexpr int Bn = 32;
constexpr int Tn = 24000;
constexpr int Fn = 8;
constexpr int Ln = 125;          // samples per chunk
constexpr int Cn = 192;          // chunks per chain (Tn / Ln)
constexpr int Sn = 5;            // timesteps per async LDS stage
constexpr int NSTG = Ln / Sn;    // 25 stages per chunk
constexpr int ROW = Sn * Fn + 4; // 44 floats: padded row -> fewer LDS bank conflicts

// workspace layout (float offsets)
constexpr size_t WS_SUM  = 0;                                   // Bn*Cn chunk phase sums
constexpr size_t WS_OFF  = WS_SUM  + (size_t)Bn * Cn;           // Bn*Cn chunk phase offsets
constexpr size_t WS_EXC  = WS_OFF  + (size_t)Bn * Cn;           // Bn*Tn excitation
constexpr size_t WS_SUMM = WS_EXC  + (size_t)Bn * Tn;           // Bn*Cn*Fn*6 affine summaries
constexpr size_t WS_INIT = WS_SUMM + (size_t)Bn * Cn * Fn * 6;  // Bn*Cn*Fn*2 chunk init states

// output layout (floats): audio (B*T), next_phase (B), next_fil (B*F*2)
constexpr int OUT_PHASE = Bn * Tn;
constexpr int OUT_FIL   = OUT_PHASE + Bn;

// ---------------- CDNA5 async global->LDS helpers ----------------
__device__ __forceinline__ void async_load_b128(unsigned lds_addr, const float* gptr) {
  unsigned long long ga = (unsigned long long)(uintptr_t)gptr;
  // GLOBAL_LOAD_ASYNC_TO_LDS_B128: vdst = LDS byte address, vaddr = 64-bit global address
  asm volatile("global_load_async_to_lds_b128 %0, %1, off"
               :: "v"(lds_addr), "v"(ga)
               : "memory");
}

template <int N>
__device__ __forceinline__ void wait_async() {
  asm volatile("s_wait_asynccnt %0" :: "i"(N) : "memory");
}

__device__ __forceinline__ void wait_ds0() {
  asm volatile("s_wait_dscnt 0" ::: "memory");
}

// ---------------- pass 1: per-chunk phase sums ----------------
__global__ void k1_chunk_phase_sums(const float* __restrict__ f0, float* __restrict__ ws) {
  int idx = blockIdx.x * blockDim.x + threadIdx.x;
  if (idx >= Bn * Cn) return;
  int b = idx / Cn, j = idx % Cn;
  const float* p = f0 + (size_t)b * Tn + (size_t)j * Ln;
  float s = 0.0f;
  for (int i = 0; i < Ln; ++i) s += p[i] * (1.0f / SR_F);
  ws[WS_SUM + idx] = s;
}

// ---------------- pass 2: per-batch scan of chunk sums ----------------
__global__ void k2_phase_scan(const float* __restrict__ phase_state,
                              float* __restrict__ ws, float* __restrict__ out) {
  int b = threadIdx.x;
  if (b >= Bn) return;
  float run = phase_state[b];
  for (int j = 0; j < Cn; ++j) {
    ws[WS_OFF + (size_t)b * Cn + j] = run;
    run += ws[WS_SUM + (size_t)b * Cn + j];
  }
  out[OUT_PHASE + b] = run; // next_phase = phase_state + full inclusive cumsum
}

// ---------------- pass 3: per-chunk affine summaries (async-staged cf/bw) ----------------
__global__ __launch_bounds__(32, 2) void k3_summaries(
    const float* __restrict__ f0, const float* __restrict__ oq,
    const float* __restrict__ vg, const float* __restrict__ ug,
    const float* __restrict__ cf, const float* __restrict__ bw,
    const float* __restrict__ nz, float* __restrict__ ws) {
  __shared__ float tile[2 * 2 * 32 * ROW]; // [buf][cf/bw][chunk-in-wave][ROW]
  const int lane = threadIdx.x;
  const int wpb  = Cn / 32;                  // wave-groups per batch
  const int b    = blockIdx.x / wpb;
  const int cb   = (blockIdx.x % wpb) * 32;  // first chunk of this wave
  const int j    = cb + lane;                // this lane's chunk
  const size_t bT = (size_t)b * Tn;
  const float* cfb = cf + (size_t)b * Tn * Fn;
  const float* bwb = bw + (size_t)b * Tn * Fn;
  float* wsExc = ws + WS_EXC;
  const unsigned lds0 = (unsigned)(uintptr_t)(void*)tile;

  auto issue = [&](int k, int buf) {
    // stage = 32 chunks x Sn timesteps x 8 filters = 40 floats/chunk -> 10 b128 per lane per array
#pragma unroll
    for (int i = 0; i < 10; ++i) {
      int v = i * 32 + lane;
      int c = v / 10;            // chunk within wave
      int r = v - c * 10;        // vec4 within that chunk's stage slice
      size_t gt = ((size_t)((cb + c) * Ln + k * Sn)) * Fn + (size_t)r * 4;
      unsigned lcf = lds0 + (unsigned)(((buf * 2 + 0) * 32 + c) * ROW + r * 4) * 4u;
      unsigned lbw = lds0 + (unsigned)(((buf * 2 + 1) * 32 + c) * ROW + r * 4) * 4u;
      async_load_b128(lcf, cfb + gt);
      async_load_b128(lbw, bwb + gt);
    }
  };

  // affine summary per filter: state' = A * state + Bv, A starts as I
  float A00[Fn], A01[Fn], A10[Fn], A11[Fn], B0[Fn], B1[Fn];
#pragma unroll
  for (int f = 0; f < Fn; ++f) {
    A00[f] = 1.0f; A01[f] = 0.0f; A10[f] = 0.0f; A11[f] = 1.0f; B0[f] = 0.0f; B1[f] = 0.0f;
  }
  float ph = ws[WS_OFF + (size_t)b * Cn + j];

  issue(0, 0);
  for (int k = 0; k < NSTG; ++k) {
    const int buf = k & 1;
    if (k + 1 < NSTG) {
      wait_ds0();               // previous reads of the other buffer fully drained
      issue(k + 1, (k + 1) & 1);
      wait_async<20>();         // in-order completion: <=20 outstanding => stage k landed
    } else {
      wait_async<0>();
    }
    const int rcf = ((buf * 2 + 0) * 32 + lane) * ROW;
    const int rbw = ((buf * 2 + 1) * 32 + lane) * ROW;
#pragma unroll
    for (int s = 0; s < Sn; ++s) {
      const int t = j * Ln + k * Sn + s;
      ph += f0[bT + t] * (1.0f / SR_F);
      float phase = ph - floorf(ph);
      float oqv = oq[bT + t];
      float tp = oqv * 0.66f;
      float tn = oqv - tp;
      float rise = 0.5f * (1.0f - __cosf(PI_F * phase / (tp + 1e-6f)));
      float fall = __cosf(PI_F * (phase - tp) / (2.0f * tn + 1e-6f));
      float wav = (phase < tp) ? rise : ((phase < oqv) ? fall : 0.0f);
      float exc = wav * vg[bT + t] + nz[bT + t] * ug[bT + t];
      wsExc[bT + t] = exc;
#pragma unroll
      for (int f = 0; f < Fn; ++f) {
        float bwv = tile[rbw + s * Fn + f];
        float cfv = tile[rcf + s * Fn + f];
        float r   = __expf((-PI_F / SR_F) * bwv);
        float th  = (2.0f * PI_F / SR_F) * cfv;
        float r2  = r * r;                         // a2
        float na1 = 2.0f * r * __cosf(th);         // -a1
        float g   = (1.0f - r2) * __sinf(th);
        float x   = exc * g;
        // M_t = [[-a1,-a2],[1,0]];  A <- M_t*A ; Bv <- M_t*Bv + [x,0]
        float n0 = fmaf(na1, A00[f], -r2 * A10[f]);
        float n1 = fmaf(na1, A01[f], -r2 * A11[f]);
        float nb = fmaf(na1, B0[f], fmaf(-r2, B1[f], x));
        A10[f] = A00[f]; A11[f] = A01[f]; A00[f] = n0; A01[f] = n1;
        B1[f] = B0[f];   B0[f] = nb;
      }
    }
  }
  float* sm = ws + WS_SUMM + ((size_t)(b * Cn + j) * Fn) * 6;
#pragma unroll
  for (int f = 0; f < Fn; ++f) {
    sm[f * 6 + 0] = A00[f]; sm[f * 6 + 1] = A01[f];
    sm[f * 6 + 2] = A10[f]; sm[f * 6 + 3] = A11[f];
    sm[f * 6 + 4] = B0[f];  sm[f * 6 + 5] = B1[f];
  }
}

// ---------------- pass 4: scan chunk summaries per (b,f) chain ----------------
__global__ void k4_chain_scan(const float* __restrict__ fil_states,
                              float* __restrict__ ws, float* __restrict__ out) {
  int idx = blockIdx.x * blockDim.x + threadIdx.x; // 0..255
  if (idx >= Bn * Fn) return;
  int b = idx >> 3, f = idx & 7;
  float y1 = fil_states[((size_t)b * Fn + f) * 2 + 0];
  float y2 = fil_states[((size_t)b * Fn + f) * 2 + 1];
  for (int j = 0; j < Cn; ++j) {
    size_t s = (size_t)(b * Cn + j) * Fn + f;
    ws[WS_INIT + s * 2 + 0] = y1;
    ws[WS_INIT + s * 2 + 1] = y2;
    const float* m = ws + WS_SUMM + s * 6;
    float n1 = fmaf(m[0], y1, fmaf(m[1], y2, m[4]));
    float n2 = fmaf(m[2], y1, fmaf(m[3], y2, m[5]));
    y1 = n1; y2 = n2;
  }
  out[OUT_FIL + (b * Fn + f) * 2 + 0] = y1; // next_fil[...,0] = y_{T-1}
  out[OUT_FIL + (b * Fn + f) * 2 + 1] = y2; // next_fil[...,1] = y_{T-2}
}

// ---------------- pass 5: replay recurrence, write audio (async-staged cf/bw) ----------------
__global__ __launch_bounds__(32, 4) void k5_replay(
    const float* __restrict__ cf, const float* __restrict__ bw,
    float* __restrict__ ws, float* __restrict__ out) {
  __shared__ float tile[2 * 2 * 32 * ROW];
  const int lane = threadIdx.x;
  const int wpb  = Cn / 32;
  const int b    = blockIdx.x / wpb;
  const int cb   = (blockIdx.x % wpb) * 32;
  const int j    = cb + lane;
  const size_t bT = (size_t)b * Tn;
  const float* cfb = cf + (size_t)b * Tn * Fn;
  const float* bwb = bw + (size_t)b * Tn * Fn;
  const float* wsExc = ws + WS_EXC;
  const unsigned lds0 = (unsigned)(uintptr_t)(void*)tile;

  auto issue = [&](int k, int buf) {
#pragma unroll
    for (int i = 0; i < 10; ++i) {
      int v = i * 32 + lane;
      int c = v / 10;
      int r = v - c * 10;
      size_t gt = ((size_t)((cb + c) * Ln + k * Sn)) * Fn + (size_t)r * 4;
      unsigned lcf = lds0 + (unsigned)(((buf * 2 + 0) * 32 + c) * ROW + r * 4) * 4u;
      unsigned lbw = lds0 + (unsigned)(((buf * 2 + 1) * 32 + c) * ROW + r * 4) * 4u;
      async_load_b128(lcf, cfb + gt);
      async_load_b128(lbw, bwb + gt);
    }
  };

  float y1[Fn], y2[Fn];
  const float* is = ws + WS_INIT + ((size_t)(b * Cn + j) * Fn) * 2;
#pragma unroll
  for (int f = 0; f < Fn; ++f) { y1[f] = is[f * 2 + 0]; y2[f] = is[f * 2 + 1]; }

  issue(0, 0);
  for (int k = 0; k < NSTG; ++k) {
    const int buf = k & 1;
    if (k + 1 < NSTG) {
      wait_ds0();
      issue(k + 1, (k + 1) & 1);
      wait_async<20>();
    } else {
      wait_async<0>();
    }
    const int rcf = ((buf * 2 + 0) * 32 + lane) * ROW;
    const int rbw = ((buf * 2 + 1) * 32 + lane) * ROW;
#pragma unroll
    for (int s = 0; s < Sn; ++s) {
      const int t = j * Ln + k * Sn + s;
      float exc = wsExc[bT + t];
      float acc = 0.0f;
#pragma unroll
      for (int f = 0; f < Fn; ++f) {
        float bwv = tile[rbw + s * Fn + f];
        float cfv = tile[rcf + s * Fn + f];
        float r   = __expf((-PI_F / SR_F) * bwv);
        float th  = (2.0f * PI_F / SR_F) * cfv;
        float r2  = r * r;
        float na1 = 2.0f * r * __cosf(th);
        float g   = (1.0f - r2) * __sinf(th);
        float x   = exc * g;
        float yt  = fmaf(na1, y1[f], fmaf(-r2, y2[f], x)); // y = x - a1*y1 - a2*y2
        acc += yt;
        y2[f] = y1[f]; y1[f] = yt;
      }
      out[bT + t] = acc;
    }
  }
}

// ---------------- host entry ----------------
extern "C" void kernel_launch(void* const* d_in, const int* in_sizes, int n_in,
                              void* d_out, int out_size, void* d_ws, size_t ws_size,
                              hipStream_t stream) {
  (void)in_sizes; (void)n_in; (void)out_size; (void)ws_size;
  const float* f0 = (const float*)d_in[0];
  const float* oq = (const float*)d_in[1];
  const float* vg = (const float*)d_in[2];
  const float* ug = (const float*)d_in[3];
  const float* cf = (const float*)d_in[4];
  const float* bw = (const float*)d_in[5];
  const float* ps = (const float*)d_in[6];
  const float* fs = (const float*)d_in[7];
  const float* nz = (const float*)d_in[8];
  float* out = (float*)d_out;
  float* ws  = (float*)d_ws;

  k1_chunk_phase_sums<<<(Bn * Cn + 255) / 256, 256, 0, stream>>>(f0, ws);
  k2_phase_scan<<<1, 32, 0, stream>>>(ps, ws, out);
  k3_summaries<<<Bn * (Cn / 32), 32, 0, stream>>>(f0, oq, vg, ug, cf, bw, nz, ws);
  k4_chain_scan<<<(Bn * Fn + 127) / 128, 128, 0, stream>>>(fs, ws, out);
  k5_replay<<<Bn * (Cn / 32), 32, 0, stream>>>(cf, bw, ws, out);
}